// Triplet_Loss_Offset_11132555231532
// MI455X (gfx1250) — compile-verified
//
#include <hip/hip_runtime.h>

#define NW 16
#define NG 5
#define NF 10
#define NPAIR 15          // NG+NF
#define STEPW 16          // 1+NG+NF
#define LSEQ 384
#define DF 64
#define NT 24             // LSEQ/16
#define NDIAG 47          // 2*NT-1
#define GAMMA_F 5.0f
#define BIGF 1.0e9f

typedef float v2f __attribute__((ext_vector_type(2)));
typedef float v8f __attribute__((ext_vector_type(8)));

// ---- dynamic LDS layout (floats) ------------------------------------------
constexpr int LDA        = 68;                         // padded row stride
constexpr int OFF_AS     = 0;                          // A staged   [384][68]
constexpr int OFF_BS     = OFF_AS + LSEQ * LDA;        // B staged   [384][68]
constexpr int OFF_NA     = OFF_BS + LSEQ * LDA;        // row norms A
constexpr int OFF_NB     = OFF_NA + LSEQ;              // row norms B
constexpr int OFF_TOP    = OFF_NB + LSEQ;              // top boundary R
constexpr int OFF_LEFT   = OFF_TOP + (LSEQ + 1);       // left boundary R
constexpr int OFF_CORN   = OFF_LEFT + (LSEQ + 1);      // corner grid 25x25
constexpr int OFF_TILE   = OFF_CORN + (NT + 1) * (NT + 1); // 16 x 16 x 17
constexpr int OFF_RES    = OFF_TILE + 16 * 16 * 17;    // endpoint value
constexpr int SMEM_FLOATS = OFF_RES + 1;
constexpr int SMEM_BYTES  = SMEM_FLOATS * 4;           // ~235 KB < 320 KB WGP LDS

// ---- CDNA5 async global->LDS copy (ASYNCcnt path), with fallback ----------
#if defined(__has_builtin)
#if __has_builtin(__builtin_amdgcn_global_load_async_to_lds_b128)
#define USE_ASYNC_COPY 1
#endif
#endif

using gv4i = int __attribute__((vector_size(16)));     // matches builtin param
#define GPTR16(x) ((gv4i*)(x))
#define LPTR16(x) ((__attribute__((address_space(3))) gv4i*)(x))

__device__ inline void stage16(const float* g, float* l) {
#ifdef USE_ASYNC_COPY
  __builtin_amdgcn_global_load_async_to_lds_b128(GPTR16(g), LPTR16(l), 0, 0);
#else
  *(float4*)l = *(const float4*)g;
#endif
}

__device__ inline void stage_wait() {
#ifdef USE_ASYNC_COPY
#if __has_builtin(__builtin_amdgcn_s_wait_asynccnt)
  __builtin_amdgcn_s_wait_asynccnt(0);
#else
  asm volatile("s_wait_asynccnt 0x0" ::: "memory");
#endif
#endif
}

// ---- one 16x16 D tile via V_WMMA_F32_16X16X4_F32 (full wave, LDS operands)
__device__ inline void wmma_tile(const float* Asm, const float* Bsm,
                                 const float* nA, const float* nB,
                                 float* tileBuf, int slot, int ti, int tj,
                                 int lane) {
  const int  fm   = lane & 15;               // A: M, B: N
  const int  koff = (lane >> 4) << 1;        // per-lane K sub-offset
  const float* aBase = Asm + (16 * ti + fm) * LDA + koff;
  const float* bBase = Bsm + (16 * tj + fm) * LDA + koff;
  v8f acc = {};
  v2f av = *(const v2f*)(aBase);
  v2f bv = *(const v2f*)(bBase);
#pragma unroll
  for (int ks = 0; ks < 16; ++ks) {
    v2f an, bn;
    if (ks < 15) {                           // prefetch next fragments
      an = *(const v2f*)(aBase + (ks + 1) * 4);
      bn = *(const v2f*)(bBase + (ks + 1) * 4);
    }
    acc = __builtin_amdgcn_wmma_f32_16x16x4_f32(
        false, av, false, bv, (short)0, acc, false, false);
    if (ks < 15) { av = an; bv = bn; }
  }
  // epilogue: D = nA + nB - 2*acc  -> LDS tile (lane owns column N, rows Mb..)
  const int Mb = (lane >> 4) << 3;           // 0 or 8
  const float  nbv = nB[16 * tj + fm];
  const float4 na0 = *(const float4*)(nA + 16 * ti + Mb);
  const float4 na1 = *(const float4*)(nA + 16 * ti + Mb + 4);
  float* tb = tileBuf + slot * 272 + Mb * 17 + fm;
  tb[0 * 17] = na0.x + nbv - 2.0f * acc[0];
  tb[1 * 17] = na0.y + nbv - 2.0f * acc[1];
  tb[2 * 17] = na0.z + nbv - 2.0f * acc[2];
  tb[3 * 17] = na0.w + nbv - 2.0f * acc[3];
  tb[4 * 17] = na1.x + nbv - 2.0f * acc[4];
  tb[5 * 17] = na1.y + nbv - 2.0f * acc[5];
  tb[6 * 17] = na1.z + nbv - 2.0f * acc[6];
  tb[7 * 17] = na1.w + nbv - 2.0f * acc[7];
}

// ---------------------------------------------------------------------------
// Kernel 1: fused  D = |a|^2+|b|^2-2ABt + tiled anti-diagonal soft-DTW DP.
// One workgroup per pair; WMMA per full wave, DP per 16-lane half-wave.
// ---------------------------------------------------------------------------
__global__ __launch_bounds__(256)
void sdtw_pair_kernel(const float* __restrict__ data,
                      const int* __restrict__ lens,
                      float* __restrict__ dist) {
  extern __shared__ float smem[];
  float* Asm     = smem + OFF_AS;
  float* Bsm     = smem + OFF_BS;
  float* nA      = smem + OFF_NA;
  float* nB      = smem + OFF_NB;
  float* tileBuf = smem + OFF_TILE;
  volatile float* topRv   = smem + OFF_TOP;
  volatile float* leftCv  = smem + OFF_LEFT;
  volatile float* cornerv = smem + OFF_CORN;
  volatile float* resv    = smem + OFF_RES;

  const int p  = blockIdx.x;
  const int w  = p / NPAIR;
  const int kp = p - w * NPAIR;
  const int ai = w * STEPW;          // anchor index
  const int oi = ai + 1 + kp;        // other index
  const float* __restrict__ A = data + (size_t)ai * LSEQ * DF;
  const float* __restrict__ B = data + (size_t)oi * LSEQ * DF;
  const int la = lens[ai];
  const int lb = lens[oi];

  const int t = threadIdx.x;

  // -- stage A,B global -> LDS via async b128 copies (16B chunks) ------------
  {
    const int chunks = LSEQ * (DF / 4);          // 6144 per matrix
    for (int idx = t; idx < chunks; idx += 256) {
      const int r  = idx >> 4;
      const int c4 = (idx & 15) << 2;
      stage16(A + r * DF + c4, Asm + r * LDA + c4);
      stage16(B + r * DF + c4, Bsm + r * LDA + c4);
    }
    stage_wait();
  }
  __syncthreads();

  // -- squared row norms (from LDS) + DP boundary init -----------------------
  for (int r = t; r < LSEQ; r += 256) {
    const float4* ar = (const float4*)(Asm + r * LDA);
    const float4* br = (const float4*)(Bsm + r * LDA);
    float sa = 0.f, sb = 0.f;
#pragma unroll
    for (int c = 0; c < DF / 4; ++c) {
      float4 x = ar[c];
      sa += x.x * x.x + x.y * x.y + x.z * x.z + x.w * x.w;
      float4 y = br[c];
      sb += y.x * y.x + y.y * y.y + y.z * y.z + y.w * y.w;
    }
    nA[r] = sa;
    nB[r] = sb;
  }
  for (int j = t; j <= LSEQ; j += 256) {
    topRv[j]  = BIGF;                  // R(0, j>=1) = BIG
    leftCv[j] = BIGF;                  // R(i>=1, 0) = BIG
  }
  if (t < NT + 1) {
    cornerv[t * (NT + 1)] = BIGF;
    cornerv[t]            = BIGF;
  }
  if (t == 0) { cornerv[0] = 0.f; resv[0] = 0.f; }   // R(0,0)=0
  __syncthreads();

  const int wave  = __builtin_amdgcn_readfirstlane(t >> 5);  // scalar wave id
  const int lane  = t & 31;
  const int half  = lane >> 4;         // which 16-lane half
  const int lr    = lane & 15;         // DP row within tile
  const float ig  = 1.0f / GAMMA_F;

  // -- tile-wavefront over 47 tile-diagonals ---------------------------------
  for (int d = 0; d < NDIAG; ++d) {
    const int tlo = (d - (NT - 1)) > 0 ? (d - (NT - 1)) : 0;
    const int thi = d < (NT - 1) ? d : (NT - 1);
    const int cnt = thi - tlo + 1;
    for (int base = 0; base < cnt; base += 16) {   // 16 half-waves per pass
      // ---- full-wave WMMA for this wave's (up to) two tiles ---------------
      // (scalar wave id -> scalar branch -> WMMA always runs with EXEC=~0)
#pragma unroll
      for (int qq = 0; qq < 2; ++qq) {
        const int idx = base + wave * 2 + qq;      // SGPR-uniform
        if (idx < cnt) {
          const int ti = tlo + idx;
          const int tj = d - ti;
          wmma_tile(Asm, Bsm, nA, nB, tileBuf, wave * 2 + qq, ti, tj, lane);
        }
      }
      __builtin_amdgcn_wave_barrier();             // order LDS write -> read

      // ---- per-half-wave DP: 31 mini-diagonals, lane lr owns row lr -------
      const int  myIdx   = base + wave * 2 + half;
      const bool hasTile = myIdx < cnt;
      const int  ti      = tlo + (hasTile ? myIdx : 0);
      const int  tj      = d - ti;
      const int  slot    = wave * 2 + half;
      const float* tb    = tileBuf + slot * 272 + lr * 17;

      float h1 = 0.f, h2 = 0.f, right = 0.f;
      float dpre = tb[0];                          // D(lr, 0) prefetched
      for (int s = 0; s < 31; ++s) {
        float u1 = __shfl_up(h1, 1);   // R(i-1, j)   from lane lr-1
        float u2 = __shfl_up(h2, 1);   // R(i-1, j-1) from lane lr-1
        const int  lc    = s - lr;
        const bool valid = hasTile && (lc >= 0) && (lc < 16);
        const float Dij  = dpre;
        int ln = s + 1 - lr;                       // prefetch next D element
        ln = ln < 0 ? 0 : (ln > 15 ? 15 : ln);
        dpre = tb[ln];
        float nv = h1;
        if (valid) {
          float rup, rdiag, rleft;
          if (lr == 0) {
            rup   = topRv[16 * tj + lc + 1];
            rdiag = (lc == 0) ? cornerv[ti * (NT + 1) + tj]
                              : topRv[16 * tj + lc];
          } else {
            rup   = u1;
            rdiag = (lc == 0) ? leftCv[16 * ti + lr] : u2;
          }
          rleft = (lc == 0) ? leftCv[16 * ti + 1 + lr] : h1;

          float m = fminf(rdiag, fminf(rup, rleft));
          float ssum = __expf((m - rdiag) * ig) +
                       __expf((m - rup)   * ig) +
                       __expf((m - rleft) * ig);
          nv = Dij + m - GAMMA_F * __logf(ssum);

          const int gi = 16 * ti + 1 + lr;
          const int gj = 16 * tj + 1 + lc;
          if (lr == 15) topRv[16 * tj + lc + 1] = nv;        // new top bndry
          if (lc == 15) right = nv;                          // new left bndry
          if (lr == 15 && lc == 15)
            cornerv[(ti + 1) * (NT + 1) + (tj + 1)] = nv;    // new corner
          if (gi == la && gj == lb) resv[0] = nv;            // endpoint
        }
        h2 = h1;
        h1 = valid ? nv : h1;
      }
      if (hasTile) leftCv[16 * ti + 1 + lr] = right;
    }
    __syncthreads();
  }

  if (t == 0) dist[p] = resv[0] / (float)(la + lb);
}

// ---------------------------------------------------------------------------
// Kernel 2: 240 distances -> scalar triplet loss (one wave).
// ---------------------------------------------------------------------------
__device__ inline float median5(float a0, float a1, float a2, float a3, float a4) {
  float a[5] = {a0, a1, a2, a3, a4};
#pragma unroll
  for (int i = 0; i < 4; ++i)
#pragma unroll
    for (int j = 0; j < 4 - i; ++j)
      if (a[j] > a[j + 1]) { float tm = a[j]; a[j] = a[j + 1]; a[j + 1] = tm; }
  return a[2];
}

__global__ __launch_bounds__(32)
void triplet_reduce_kernel(const float* __restrict__ dist,
                           float* __restrict__ out) {
  const int lane = threadIdx.x;
  float contrib = 0.f;
  if (lane < NW) {
    float dg[NG], dn[NF];
#pragma unroll
    for (int i = 0; i < NG; ++i) dg[i] = dist[lane * NPAIR + i];
#pragma unroll
    for (int i = 0; i < NF; ++i) dn[i] = dist[lane * NPAIR + NG + i];

    float lk = 0.f;
    int   nz = 0;
#pragma unroll
    for (int i = 0; i < NG; ++i)
#pragma unroll
      for (int j = 0; j < NF; ++j) {
        float x = dg[i] + 1.0f - dn[j];   // MARGIN = 1
        if (x > 0.f) { lk += x; ++nz; }
      }
    float lv = lk / (1.0f + (float)nz);
    float sum_g = dg[0] + dg[1] + dg[2] + dg[3] + dg[4];
    float only_pos = sum_g * (0.01f / (float)NG);   // LAM/NG
    float mg = median5(dg[0], dg[1], dg[2], dg[3], dg[4]);
    float ms = median5(dn[0], dn[1], dn[2], dn[3], dn[4]);
    int err = 0;
#pragma unroll
    for (int i = 0; i < 5; ++i) {
      if (dg[i] < mg && dn[i] < mg) ++err;
      if (dg[i] > ms && dn[i] > ms) ++err;
      if (dg[i] < mg && dn[5 + i] < mg) ++err;
      if (dg[i] > ms && dn[5 + i] > ms) ++err;
    }
    contrib = lv + only_pos + (float)err / (float)(NF * NG);  // ALPHA = 1
  }
#pragma unroll
  for (int off = 16; off > 0; off >>= 1)
    contrib += __shfl_down(contrib, off);
  if (lane == 0) out[0] = contrib * (1.0f / (float)NW);
}

// ---------------------------------------------------------------------------
extern "C" void kernel_launch(void* const* d_in, const int* in_sizes, int n_in,
                              void* d_out, int out_size, void* d_ws, size_t ws_size,
                              hipStream_t stream) {
  const float* data = (const float*)d_in[0];
  const int*   lens = (const int*)d_in[1];
  float*       out  = (float*)d_out;
  float*       dist = (float*)d_ws;     // 240 floats of scratch

  (void)hipFuncSetAttribute(reinterpret_cast<const void*>(sdtw_pair_kernel),
                            hipFuncAttributeMaxDynamicSharedMemorySize,
                            SMEM_BYTES);
  sdtw_pair_kernel<<<NW * NPAIR, 256, SMEM_BYTES, stream>>>(data, lens, dist);
  triplet_reduce_kernel<<<1, 32, 0, stream>>>(dist, out);
}